// VecDGCNN_att_81999515615323
// MI455X (gfx1250) — compile-verified
//
#include <hip/hip_runtime.h>

#define BSZ 8
#define NPTS 2048
#define KNN_ 16
#define MC_ 16
#define SLOPE_ 0.2f
#define EPSV 1e-12f

typedef __attribute__((ext_vector_type(2))) float v2f;
typedef __attribute__((ext_vector_type(8))) float v8f;

// ---------------------------------------------------------------------------
// fp32 WMMA wrapper: D = A(16x4) x B(4x16) + C(16x16), wave32.
// A frag (v2f): lane<16 -> row=lane, K=k0,k0+1 ; lane>=16 -> row=lane-16, K=k0+2,k0+3
// B frag (v2f): lane<16 -> col=lane, K=k0,k0+1 ; lane>=16 -> col=lane-16, K=k0+2,k0+3
// C/D (v8f): vgpr r: lanes0-15 -> row r, col lane ; lanes16-31 -> row r+8, col lane-16
// ---------------------------------------------------------------------------
__device__ inline v8f wmma_f32(v2f a, v2f b, v8f c) {
#if defined(__AMDGCN__) && __has_builtin(__builtin_amdgcn_wmma_f32_16x16x4_f32)
  return __builtin_amdgcn_wmma_f32_16x16x4_f32(false, a, false, b, (short)0, c,
                                               false, false);
#else
  return c; // host pass / unsupported target: dead path
#endif
}

// ---------------------------------------------------------------------------
// Dense GEMM: Y[b][o][m] = sum_c W[o][c] * X[b][c][m]
// grid: (ceil(M/64), ceil(O/16), BSZ), block (32,4)
// Fast path (tile-exact shapes): unguarded, aligned v2f weight loads,
// unrolled so loads of later K-steps overlap the WMMA chain.
// ---------------------------------------------------------------------------
__global__ __launch_bounds__(128)
void gemm_dense_f32(const float* __restrict__ W, const float* __restrict__ X,
                    float* __restrict__ Y, int O, int Cin, int M) {
  const int lane = threadIdx.x;
  const int half = lane >> 4;
  const int l16 = lane & 15;
  const int tm = blockIdx.x * 4 + threadIdx.y;
  const int to = blockIdx.y;
  const int b = blockIdx.z;
  if (tm * 16 >= M) return;
  const int row = to * 16 + l16;
  const int col = tm * 16 + l16;
  const float* Xb = X + (size_t)b * Cin * M;
  float* Yb = Y + (size_t)b * O * M;
  v8f acc = {};
  const bool fast = ((O & 15) == 0) && ((Cin & 3) == 0) && ((M & 15) == 0);
  if (fast) {
    const float* wrow = W + (size_t)row * Cin + half * 2;
    const float* xcol = Xb + col;
#pragma unroll 4
    for (int k0 = 0; k0 < Cin; k0 += 4) {
      if ((k0 & 63) == 0) __builtin_prefetch(wrow + k0 + 64, 0, 1);
      const int ka = k0 + half * 2;
      const v2f af = *(const v2f*)(wrow + k0);
      v2f bf;
      bf.x = xcol[(size_t)ka * M];
      bf.y = xcol[(size_t)(ka + 1) * M];
      acc = wmma_f32(af, bf, acc);
    }
#pragma unroll
    for (int r = 0; r < 8; ++r) {
      const int orow = to * 16 + half * 8 + r;
      Yb[(size_t)orow * M + col] = acc[r];
    }
  } else {
    for (int k0 = 0; k0 < Cin; k0 += 4) {
      const int ka = k0 + half * 2;
      v2f af, bf;
      af.x = (row < O && ka < Cin) ? W[(size_t)row * Cin + ka] : 0.f;
      af.y = (row < O && ka + 1 < Cin) ? W[(size_t)row * Cin + ka + 1] : 0.f;
      bf.x = (col < M && ka < Cin) ? Xb[(size_t)ka * M + col] : 0.f;
      bf.y = (col < M && ka + 1 < Cin) ? Xb[(size_t)(ka + 1) * M + col] : 0.f;
      acc = wmma_f32(af, bf, acc);
    }
    if (col < M) {
#pragma unroll
      for (int r = 0; r < 8; ++r) {
        const int orow = to * 16 + half * 8 + r;
        if (orow < O) Yb[(size_t)orow * M + col] = acc[r];
      }
    }
  }
}

// ---------------------------------------------------------------------------
// Edge GEMM: B-operand synthesized on the fly from graph features:
//   y[c, (p,nl,kk)] = c<chalf ? srcF[c][p][idx] - dstF[c][p][n]
//                             : dstF[c-chalf][p][n]
// Columns m = (p * nc + nl) * 16 + kk, n = n0 + nl, M = 3*nc*16 (mult of 64).
// O mult of 16, Cin = 2*chalf mult of 8 -> fully unguarded.
// (chalf mult of 4 and ka even => ka<chalf <=> ka+1<chalf, wave-uniform.)
// ---------------------------------------------------------------------------
__global__ __launch_bounds__(128)
void gemm_edge_f32(const float* __restrict__ W, const float* __restrict__ SF,
                   const float* __restrict__ DF, const int* __restrict__ IDX,
                   float* __restrict__ Y, int O, int chalf, int nsrc, int ndst,
                   int n0, int nc) {
  const int Cin = 2 * chalf;
  const int lane = threadIdx.x;
  const int half = lane >> 4;
  const int l16 = lane & 15;
  const int tm = blockIdx.x * 4 + threadIdx.y;
  const int to = blockIdx.y;
  const int b = blockIdx.z;
  const int M = 3 * nc * KNN_;
  if (tm * 16 >= M) return;
  const int row = to * 16 + l16;
  const int col = tm * 16 + l16;
  const int p = col / (nc * KNN_);
  const int r_ = col - p * nc * KNN_;
  const int nl = r_ >> 4;
  const int kk = r_ & 15;
  const int nn = n0 + nl;
  const int sidx = IDX[((size_t)b * ndst + nn) * KNN_ + kk];
  const float* sb = SF + (size_t)b * chalf * 3 * nsrc + (size_t)p * nsrc + sidx;
  const float* db = DF + (size_t)b * chalf * 3 * ndst + (size_t)p * ndst + nn;
  const float* wrow = W + (size_t)row * Cin + half * 2;
  v8f acc = {};
#pragma unroll 2
  for (int k0 = 0; k0 < Cin; k0 += 4) {
    const int ka = k0 + half * 2;
    const v2f af = *(const v2f*)(wrow + k0);
    v2f bf;
    if (ka < chalf) {
      bf.x = sb[(size_t)ka * 3 * nsrc] - db[(size_t)ka * 3 * ndst];
      bf.y = sb[(size_t)(ka + 1) * 3 * nsrc] - db[(size_t)(ka + 1) * 3 * ndst];
    } else {
      bf.x = db[(size_t)(ka - chalf) * 3 * ndst];
      bf.y = db[(size_t)(ka + 1 - chalf) * 3 * ndst];
    }
    acc = wmma_f32(af, bf, acc);
  }
  float* Yb = Y + (size_t)b * O * M;
#pragma unroll
  for (int r = 0; r < 8; ++r) {
    const int orow = to * 16 + half * 8 + r;
    Yb[(size_t)orow * M + col] = acc[r];
  }
}

// ---------------------------------------------------------------------------
// Layer-0 edge features (cross-product mode, cin=3, src==dst==x):
//   y0 rows: 0 = cross(normalize(x_n), x_idx), 1 = x_idx - x_n, 2 = x_n
// ---------------------------------------------------------------------------
__global__ void build_y0_kernel(const float* __restrict__ X,
                                const int* __restrict__ IDX,
                                float* __restrict__ Y, int nsrc, int ndst,
                                int n0, int nc) {
  const int total = BSZ * 3 * 3 * nc * KNN_;
  int tid = blockIdx.x * 256 + threadIdx.x;
  if (tid >= total) return;
  const int kk = tid % KNN_;
  const int nl = (tid / KNN_) % nc;
  const int p = (tid / (KNN_ * nc)) % 3;
  const int cch = (tid / (KNN_ * nc * 3)) % 3;
  const int b = tid / (KNN_ * nc * 9);
  const int n = n0 + nl;
  const float* xb = X + (size_t)b * 3 * nsrc;
  const int si = IDX[((size_t)b * ndst + n) * KNN_ + kk];
  const float a0 = xb[si], a1 = xb[nsrc + si], a2 = xb[2 * nsrc + si];
  const float d0 = xb[n], d1 = xb[nsrc + n], d2 = xb[2 * nsrc + n];
  float val;
  if (cch == 2) {
    val = (p == 0) ? d0 : (p == 1) ? d1 : d2;
  } else if (cch == 1) {
    val = ((p == 0) ? a0 : (p == 1) ? a1 : a2) -
          ((p == 0) ? d0 : (p == 1) ? d1 : d2);
  } else {
    const float dn = sqrtf(d0 * d0 + d1 * d1 + d2 * d2);
    const float iv = 1.f / fmaxf(dn, EPSV);
    const float x0 = d0 * iv, x1 = d1 * iv, x2 = d2 * iv;
    const float c0 = x1 * a2 - x2 * a1;
    const float c1 = x2 * a0 - x0 * a2;
    const float c2 = x0 * a1 - x1 * a0;
    val = (p == 0) ? c0 : (p == 1) ? c1 : c2;
  }
  Y[((size_t)(b * 3 + cch)) * (3 * nc * KNN_) + ((size_t)p * nc + nl) * KNN_ +
    kk] = val;
}

// ---------------------------------------------------------------------------
// vec_lna epilogue: Out = t + (leaky(p)-p) * d,  d = normalize(draw over 3-axis)
// T/Out: (b,O,3,L), Draw: (b,Od,3,L) (Od == O or 1)
// ---------------------------------------------------------------------------
__global__ void lna_combine(const float* __restrict__ T,
                            const float* __restrict__ Dr,
                            float* __restrict__ Out, int O, int Od, int L) {
  const int total = BSZ * O * L;
  int tid = blockIdx.x * 256 + threadIdx.x;
  if (tid >= total) return;
  const int l = tid % L;
  const int o = (tid / L) % O;
  const int b = tid / (L * O);
  const int od = (Od == O) ? o : 0;
  const float* dbp = Dr + (((size_t)b * Od + od) * 3) * L + l;
  float d0 = dbp[0], d1 = dbp[(size_t)L], d2 = dbp[(size_t)2 * L];
  const float nrm = sqrtf(d0 * d0 + d1 * d1 + d2 * d2);
  const float inv = 1.f / fmaxf(nrm, EPSV);
  d0 *= inv; d1 *= inv; d2 *= inv;
  const float* tbp = T + (((size_t)b * O + o) * 3) * L + l;
  const float t0 = tbp[0], t1 = tbp[(size_t)L], t2 = tbp[(size_t)2 * L];
  const float pd = t0 * d0 + t1 * d1 + t2 * d2;
  const float cf = (pd < 0.f) ? (SLOPE_ - 1.f) * pd : 0.f;
  float* obp = Out + (((size_t)b * O + o) * 3) * L + l;
  obp[0] = t0 + cf * d0;
  obp[(size_t)L] = t1 + cf * d1;
  obp[(size_t)2 * L] = t2 + cf * d2;
}

// ---------------------------------------------------------------------------
// cevn: X (b,C,3,L) -> X * (n/max(n,eps)) / max(||n||_C, eps)
// ---------------------------------------------------------------------------
__global__ void zero_f(float* p, int n) {
  int i = blockIdx.x * 256 + threadIdx.x;
  if (i < n) p[i] = 0.f;
}
__global__ void cevn_norm(const float* __restrict__ X, float* __restrict__ Nb,
                          float* __restrict__ Sb, int C, int L) {
  const int total = BSZ * C * L;
  int tid = blockIdx.x * 256 + threadIdx.x;
  if (tid >= total) return;
  const int l = tid % L;
  const int c = (tid / L) % C;
  const int b = tid / (L * C);
  const size_t base = (((size_t)b * C + c) * 3) * L + l;
  const float x0 = X[base], x1 = X[base + L], x2 = X[base + 2 * (size_t)L];
  const float nn = sqrtf(x0 * x0 + x1 * x1 + x2 * x2);
  Nb[((size_t)b * C + c) * L + l] = nn;
  atomicAdd(&Sb[(size_t)b * L + l], nn * nn);
}
__global__ void cevn_scale(float* __restrict__ X, const float* __restrict__ Nb,
                           const float* __restrict__ Sb, int C, int L) {
  const int total = BSZ * C * L;
  int tid = blockIdx.x * 256 + threadIdx.x;
  if (tid >= total) return;
  const int l = tid % L;
  const int c = (tid / L) % C;
  const int b = tid / (L * C);
  const float nn = Nb[((size_t)b * C + c) * L + l];
  const float s = sqrtf(Sb[(size_t)b * L + l]);
  const float f = (nn / fmaxf(nn, EPSV)) / fmaxf(s, EPSV);
  const size_t base = (((size_t)b * C + c) * 3) * L + l;
  X[base] *= f;
  X[base + L] *= f;
  X[base + 2 * (size_t)L] *= f;
}

// ---------------------------------------------------------------------------
// FPS: one block per batch, iterative argmax in LDS (first-index tiebreak).
// ---------------------------------------------------------------------------
__global__ void fps_kernel(const float* __restrict__ X, int* __restrict__ out,
                           int n, int K) {
  __shared__ float mind[NPTS];
  __shared__ float rv[256];
  __shared__ int ri[256];
  const int b = blockIdx.x;
  const int t = threadIdx.x;
  const float* xb = X + (size_t)b * 3 * n;
  const float p0x = xb[0], p0y = xb[n], p0z = xb[2 * n];
  for (int j = t; j < n; j += 256) {
    const float dx = xb[j] - p0x, dy = xb[n + j] - p0y, dz = xb[2 * n + j] - p0z;
    mind[j] = dx * dx + dy * dy + dz * dz;
  }
  if (t == 0) out[(size_t)b * K] = 0;
  __syncthreads();
  for (int s = 1; s < K; ++s) {
    float bv = -1.f;
    int bidx = 0;
    for (int j = t; j < n; j += 256) {
      const float v = mind[j];
      if (v > bv) { bv = v; bidx = j; }
    }
    rv[t] = bv; ri[t] = bidx;
    __syncthreads();
    for (int off = 128; off > 0; off >>= 1) {
      if (t < off) {
        if (rv[t + off] > rv[t] ||
            (rv[t + off] == rv[t] && ri[t + off] < ri[t])) {
          rv[t] = rv[t + off];
          ri[t] = ri[t + off];
        }
      }
      __syncthreads();
    }
    const int nid = ri[0];
    if (t == 0) out[(size_t)b * K + s] = nid;
    const float qx = xb[nid], qy = xb[n + nid], qz = xb[2 * n + nid];
    for (int j = t; j < n; j += 256) {
      const float dx = xb[j] - qx, dy = xb[n + j] - qy, dz = xb[2 * n + j] - qz;
      const float d = dx * dx + dy * dy + dz * dz;
      if (d < mind[j]) mind[j] = d;
    }
    __syncthreads();
  }
}

// ---------------------------------------------------------------------------
// Brute-force KNN-16 (smallest d2, stable tie order == top_k semantics).
// SF: (b,c,3,nsrc)  DF: (b,c,3,ndst)
// ---------------------------------------------------------------------------
__global__ void knn_kernel(const float* __restrict__ SF,
                           const float* __restrict__ DF, int* __restrict__ idx,
                           int c, int nsrc, int ndst) {
  const int nd = blockIdx.x * blockDim.x + threadIdx.x;
  const int b = blockIdx.y;
  if (nd >= ndst) return;
  const float* sb = SF + (size_t)b * c * 3 * nsrc;
  const float* db = DF + (size_t)b * c * 3 * ndst;
  float bd[KNN_];
  int bi[KNN_];
#pragma unroll
  for (int q = 0; q < KNN_; ++q) { bd[q] = 3.4e38f; bi[q] = 0; }
  const int dim = c * 3;
  for (int ns = 0; ns < nsrc; ++ns) {
    float d2 = 0.f;
    for (int f = 0; f < dim; ++f) {
      const float df = db[(size_t)f * ndst + nd] - sb[(size_t)f * nsrc + ns];
      d2 += df * df;
    }
    if (d2 < bd[KNN_ - 1]) {
#pragma unroll
      for (int q = KNN_ - 1; q >= 0; --q) {
        if (d2 < bd[q]) {
          const bool shift = (q > 0) && (d2 < bd[q - 1]);
          const float nv = shift ? bd[q - 1] : d2;
          const int ni = shift ? bi[q - 1] : ns;
          bd[q] = nv;
          bi[q] = ni;
        }
      }
    }
  }
#pragma unroll
  for (int q = 0; q < KNN_; ++q) idx[((size_t)b * ndst + nd) * KNN_ + q] = bi[q];
}

// ---------------------------------------------------------------------------
// Attention: per (b, nl) block. qk -> head sums -> softmax(k) -> weighted vf.
// ---------------------------------------------------------------------------
__global__ void attention_kernel(const float* __restrict__ KF,
                                 const float* __restrict__ QFp,
                                 const float* __restrict__ VF,
                                 float* __restrict__ OutF, int co, int nc,
                                 int n0, int ndst) {
  __shared__ float s_h[512];
  __shared__ float s_att[512];
  const int nl = blockIdx.x, b = blockIdx.y;
  const int nh = co / MC_;
  const int t = threadIdx.x;
  for (int q = t; q < nh * 16; q += 256) s_h[q] = 0.f;
  __syncthreads();
  const int n = n0 + nl;
  const size_t kb = (size_t)b * co * 3 * nc * KNN_;
  const size_t qb = (size_t)b * co * 3 * ndst;
  for (int ok = t; ok < co * 16; ok += 256) {
    const int kk = ok & 15, o = ok >> 4;
    float s = 0.f;
#pragma unroll
    for (int p = 0; p < 3; ++p)
      s += KF[kb + (((size_t)o * 3 + p) * nc + nl) * KNN_ + kk] *
           QFp[qb + ((size_t)o * 3 + p) * ndst + n];
    atomicAdd(&s_h[(o >> 4) * 16 + kk], s);
  }
  __syncthreads();
  if (t < nh) {
    float v[16];
    float mx = -3.4e38f;
#pragma unroll
    for (int kk = 0; kk < 16; ++kk) {
      v[kk] = s_h[t * 16 + kk] * 0.14433756729740646f; // 1/sqrt(48)
      mx = fmaxf(mx, v[kk]);
    }
    float sm = 0.f;
#pragma unroll
    for (int kk = 0; kk < 16; ++kk) { v[kk] = expf(v[kk] - mx); sm += v[kk]; }
    const float is = 1.f / sm;
#pragma unroll
    for (int kk = 0; kk < 16; ++kk) s_att[t * 16 + kk] = v[kk] * is;
  }
  __syncthreads();
  for (int op = t; op < co * 3; op += 256) {
    const int p = op % 3, o = op / 3;
    const float* at = &s_att[(o >> 4) * 16];
    float acc = 0.f;
#pragma unroll
    for (int kk = 0; kk < 16; ++kk)
      acc += at[kk] * VF[kb + (((size_t)o * 3 + p) * nc + nl) * KNN_ + kk];
    OutF[qb + ((size_t)o * 3 + p) * ndst + n] = acc;
  }
}

// ---------------------------------------------------------------------------
// small elementwise kernels
// ---------------------------------------------------------------------------
__global__ void meank_kernel(const float* __restrict__ T, float* __restrict__ O,
                             int co, int nc, int n0, int ndst) {
  const int total = BSZ * co * 3 * nc;
  int tid = blockIdx.x * 256 + threadIdx.x;
  if (tid >= total) return;
  const int nl = tid % nc;
  const int p = (tid / nc) % 3;
  const int o = (tid / (nc * 3)) % co;
  const int b = tid / (nc * 3 * co);
  const float* tb = T + ((((size_t)b * co + o) * 3 + p) * nc + nl) * KNN_;
  float s = 0.f;
#pragma unroll
  for (int kk = 0; kk < KNN_; ++kk) s += tb[kk];
  O[(((size_t)b * co + o) * 3 + p) * ndst + n0 + nl] = s * (1.f / 16.f);
}
__global__ void mean_n_kernel(const float* __restrict__ F, float* __restrict__ O,
                              int C, int n) {
  const int total = BSZ * C * 3;
  int tid = blockIdx.x * 256 + threadIdx.x;
  if (tid >= total) return;
  const int p = tid % 3;
  const int c = (tid / 3) % C;
  const int b = tid / (3 * C);
  const float* fb = F + (((size_t)b * C + c) * 3 + p) * n;
  float s = 0.f;
  for (int j = 0; j < n; ++j) s += fb[j];
  O[((size_t)b * C + c) * 3 + p] = s / (float)n;
}
__global__ void concat_kernel(const float* __restrict__ F,
                              const float* __restrict__ G,
                              float* __restrict__ CC, int co, int n) {
  const int total = BSZ * 2 * co * 3 * n;
  int tid = blockIdx.x * 256 + threadIdx.x;
  if (tid >= total) return;
  const int j = tid % n;
  const int p = (tid / n) % 3;
  const int c2 = (tid / (n * 3)) % (2 * co);
  const int b = tid / (n * 3 * 2 * co);
  const float v = (c2 < co) ? F[(((size_t)b * co + c2) * 3 + p) * n + j]
                            : G[((size_t)b * co + (c2 - co)) * 3 + p];
  CC[tid] = v;
}
__global__ void gather_xyz_kernel(const float* __restrict__ X,
                                  const int* __restrict__ fi,
                                  float* __restrict__ O, int nsrc, int ndst) {
  const int total = BSZ * 3 * ndst;
  int tid = blockIdx.x * 256 + threadIdx.x;
  if (tid >= total) return;
  const int j = tid % ndst;
  const int p = (tid / ndst) % 3;
  const int b = tid / (3 * ndst);
  O[tid] = X[((size_t)b * 3 + p) * nsrc + fi[(size_t)b * ndst + j]];
}
__global__ void gather_feat_kernel(const float* __restrict__ F,
                                   const int* __restrict__ fi,
                                   float* __restrict__ O, int C, int nsrc,
                                   int ndst) {
  const int total = BSZ * C * 3 * ndst;
  int tid = blockIdx.x * 256 + threadIdx.x;
  if (tid >= total) return;
  const int j = tid % ndst;
  const int p = (tid / ndst) % 3;
  const int c = (tid / (ndst * 3)) % C;
  const int b = tid / (ndst * 3 * C);
  O[tid] = F[(((size_t)b * C + c) * 3 + p) * nsrc + fi[(size_t)b * ndst + j]];
}
__global__ void copy_kernel(const float* a, float* o, int n) {
  int i = blockIdx.x * 256 + threadIdx.x;
  if (i < n) o[i] = a[i];
}
__global__ void scale_out_kernel(const float* __restrict__ FM,
                                 float* __restrict__ out) {
  const int b = threadIdx.x;
  if (b >= BSZ) return;
  const float* f = FM + (size_t)b * 256 * 3;
  float s = 0.f;
  for (int c = 0; c < 256; ++c) {
    const float x0 = f[c * 3], x1 = f[c * 3 + 1], x2 = f[c * 3 + 2];
    s += sqrtf(x0 * x0 + x1 * x1 + x2 * x2);
  }
  out[b] = s * (640.0f / 256.0f);
}
__global__ void write_final_kernel(const float* __restrict__ ZS,
                                   const float* __restrict__ ZI,
                                   float* __restrict__ out) {
  int tid = blockIdx.x * 256 + threadIdx.x;
  if (tid >= BSZ * 256) return;
  const int c = tid % 256;
  const int b = tid / 256;
  const float* zs = ZS + ((size_t)(b * 256 + c)) * 3;
  const float* zi = ZI + ((size_t)(b * 256 + c)) * 3;
  float v = 0.f;
#pragma unroll
  for (int p = 0; p < 3; ++p) {
    out[8 + (size_t)(b * 256 + c) * 3 + p] = zs[p];
    v += zi[p] * zs[p];
  }
  out[8 + 6144 + b * 256 + c] = v;
}

// ===========================================================================
extern "C" void kernel_launch(void* const* d_in, const int* in_sizes, int n_in,
                              void* d_out, int out_size, void* d_ws,
                              size_t ws_size, hipStream_t stream) {
  (void)out_size;
  static const int FEATS[8] = {32, 32, 64, 64, 128, 256, 512, 512};
  static const int CINL[8] = {3, 64, 64, 128, 128, 256, 512, 1024};
  static const int NSRC[8] = {2048, 2048, 2048, 512, 512, 128, 128, 32};
  static const int NDST[8] = {2048, 2048, 512, 512, 128, 128, 32, 32};
  static const int CSRC[8] = {1, 32, 32, 64, 64, 128, 256, 512};
  static const int NCHK[8] = {512, 512, 512, 512, 128, 128, 32, 32};
  static const int ISDS[8] = {0, 0, 1, 0, 1, 0, 1, 0};
  if (n_in < 56 || !d_out || !d_ws) return;

  // ---- input mapping (try dict-insertion order, then alphabetical pytree) ----
  struct WMap {
    int x, vl[8], vd[8], ql[6], qd[6], kl[6], kd[6], gl[6], gd[6], cw, cd, fc;
  };
  long esz[64];
  auto fill = [&](bool alpha) -> WMap {
    WMap mm{};
    int p = 0;
    auto put = [&](int& s, long e) { s = p; esz[p++] = e; };
    auto putV = [&](WMap& w) {
      for (int i = 0; i < 8; ++i) {
        put(w.vl[i], (long)FEATS[i] * CINL[i]);
        put(w.vd[i], (long)FEATS[i] * FEATS[i]);
      }
    };
    auto putQ = [&](WMap& w) {
      for (int j = 0; j < 6; ++j) {
        int i = j + 2;
        put(w.ql[j], (long)FEATS[i] * FEATS[i - 1]);
        put(w.qd[j], (long)FEATS[i] * FEATS[i]);
      }
    };
    auto putK = [&](WMap& w) {
      for (int j = 0; j < 6; ++j) {
        int i = j + 2;
        put(w.kl[j], (long)FEATS[i] * 2 * FEATS[i - 1]);
        put(w.kd[j], (long)FEATS[i] * FEATS[i]);
      }
    };
    auto putG = [&](WMap& w) {
      for (int j = 0; j < 6; ++j) {
        int i = j + 2;
        put(w.gl[j], (long)FEATS[i] * 2 * FEATS[i]);
        put(w.gd[j], (long)FEATS[i] * FEATS[i]);
      }
    };
    if (!alpha) {
      put(mm.x, (long)BSZ * 3 * NPTS);
      putV(mm); putQ(mm); putK(mm); putG(mm);
      put(mm.cw, 256L * 512); put(mm.cd, 256); put(mm.fc, 256L * 256);
    } else {
      putG(mm); putK(mm); putQ(mm); putV(mm);
      put(mm.cw, 256L * 512); put(mm.cd, 256); put(mm.fc, 256L * 256);
      put(mm.x, (long)BSZ * 3 * NPTS);
    }
    return mm;
  };
  auto okm = [&]() {
    for (int t = 0; t < 56; ++t)
      if (esz[t] != (long)in_sizes[t]) return false;
    return true;
  };
  WMap A = fill(false);
  bool okA = okm();
  WMap Bm = fill(true);
  bool okB = okm();
  WMap MP = okA ? A : (okB ? Bm : A);

  const float* xin = (const float*)d_in[MP.x];
  const float *VLp[8], *VDp[8], *QLp[6], *QDp[6], *KLp[6], *KDp[6], *GLp[6],
      *GDp[6];
  for (int i = 0; i < 8; ++i) {
    VLp[i] = (const float*)d_in[MP.vl[i]];
    VDp[i] = (const float*)d_in[MP.vd[i]];
  }
  for (int j = 0; j < 6; ++j) {
    QLp[j] = (const float*)d_in[MP.ql[j]];
    QDp[j] = (const float*)d_in[MP.qd[j]];
    KLp[j] = (const float*)d_in[MP.kl[j]];
    KDp[j] = (const float*)d_in[MP.kd[j]];
    GLp[j] = (const float*)d_in[MP.gl[j]];
    GDp[j] = (const float*)d_in[MP.gd[j]];
  }
  const float* CW = (const float*)d_in[MP.cw];
  const float* CD = (const float*)d_in[MP.cd];
  const float* FCW = (const float*)d_in[MP.fc];

  // ---- workspace carve-up ----
  char* w = (char*)d_ws;
  size_t used = 0;
  auto alloc = [&](size_t nfl) -> float* {
    float* p = (float*)(w + used);
    used += ((nfl * 4 + 255) / 256) * 256;
    return p;
  };
  float* Tb = alloc(12582912);
  float* Db = alloc(12582912);
  float* Ub = alloc(12582912);
  float* NB = alloc(4194304);
  float* SB = alloc(65536);
  float* QF = alloc(786432);
  float* FA = alloc(1572864);
  float* FB = alloc(1572864);
  float* FC = alloc(1572864);
  float* XA = alloc(49152);
  float* XB = alloc(49152);
  float* GB = alloc(12288);
  float* FF = alloc(196608);
  float* FM = alloc(6144);
  float* ZS = alloc(6144);
  float* ZI = alloc(6144);
  int* IDXB = (int*)alloc(262144);
  int* FPSI = (int*)alloc(4096);
  if (ws_size < used) return;

  auto grid1 = [](int total) { return dim3((unsigned)((total + 255) / 256)); };
  auto gemm = [&](const float* W_, const float* X_, float* Y_, int O, int Cin,
                  int M) {
    dim3 g((unsigned)((M + 63) / 64), (unsigned)((O + 15) / 16), BSZ);
    gemm_dense_f32<<<g, dim3(32, 4, 1), 0, stream>>>(W_, X_, Y_, O, Cin, M);
  };
  auto egemm = [&](const float* W_, const float* SF, const float* DF,
                   const int* ix, float* Y_, int O, int chalf, int nsrc,
                   int ndst, int n0, int nc) {
    const int M = 3 * nc * KNN_;
    dim3 g((unsigned)(M / 64), (unsigned)(O / 16), BSZ);
    gemm_edge_f32<<<g, dim3(32, 4, 1), 0, stream>>>(W_, SF, DF, ix, Y_, O,
                                                    chalf, nsrc, ndst, n0, nc);
  };
  auto combine = [&](const float* T_, const float* D_, float* O_, int O, int Od,
                     int L) {
    lna_combine<<<grid1(BSZ * O * L), 256, 0, stream>>>(T_, D_, O_, O, Od, L);
  };
  auto cevn = [&](float* X_, int C, int L) {
    zero_f<<<grid1(BSZ * L), 256, 0, stream>>>(SB, BSZ * L);
    cevn_norm<<<grid1(BSZ * C * L), 256, 0, stream>>>(X_, NB, SB, C, L);
    cevn_scale<<<grid1(BSZ * C * L), 256, 0, stream>>>(X_, NB, SB, C, L);
  };
  auto vec_lna_dense = [&](const float* Wl, const float* Wd, const float* X_,
                           float* O_, int O, int Od, int Cin, int M) {
    gemm(Wl, X_, Tb, O, Cin, M);
    gemm(Wd, Tb, Db, Od, O, M);
    combine(Tb, Db, O_, O, Od, M / 3);
  };

  // ---- layer pipeline ----
  const float* curX = xin;
  const float* curF = xin;
  float* fping[2] = {FA, FB};
  float* xping[2] = {XA, XB};
  int f_i = 0, x_i = 0;

  for (int i = 0; i < 8; ++i) {
    const int nsrc = NSRC[i], ndst = NDST[i], c = CSRC[i], co = FEATS[i];
    const float *dF, *dX;
    if (ISDS[i]) {
      fps_kernel<<<BSZ, 256, 0, stream>>>(curX, FPSI, nsrc, ndst);
      gather_xyz_kernel<<<grid1(BSZ * 3 * ndst), 256, 0, stream>>>(
          curX, FPSI, xping[x_i], nsrc, ndst);
      gather_feat_kernel<<<grid1(BSZ * c * 3 * ndst), 256, 0, stream>>>(
          curF, FPSI, FC, c, nsrc, ndst);
      dX = xping[x_i];
      x_i ^= 1;
      dF = FC;
    } else {
      dX = curX;
      dF = curF;
    }
    knn_kernel<<<dim3((unsigned)((ndst + 127) / 128), BSZ), 128, 0, stream>>>(
        curF, dF, IDXB, c, nsrc, ndst);
    float* outF = fping[f_i];
    if (i >= 2) {
      const int j = i - 2;
      vec_lna_dense(QLp[j], QDp[j], dF, QF, co, co, c, 3 * ndst);
      cevn(QF, co, ndst);
    }
    const int nc = NCHK[i];
    for (int n0 = 0; n0 < ndst; n0 += nc) {
      const int M = 3 * nc * KNN_;
      if (i == 0) {
        build_y0_kernel<<<grid1(BSZ * 9 * nc * KNN_), 256, 0, stream>>>(
            curF, IDXB, Ub, nsrc, ndst, n0, nc);
        gemm(VLp[0], Ub, Tb, co, 3, M);
      } else {
        egemm(VLp[i], curF, dF, IDXB, Tb, co, c, nsrc, ndst, n0, nc);
      }
      gemm(VDp[i], Tb, Db, co, co, M);
      combine(Tb, Db, Tb, co, co, M / 3);
      if (i < 2) {
        meank_kernel<<<grid1(BSZ * co * 3 * nc), 256, 0, stream>>>(
            Tb, outF, co, nc, n0, ndst);
      } else {
        const int j = i - 2;
        egemm(KLp[j], curF, dF, IDXB, Ub, co, c, nsrc, ndst, n0, nc);
        gemm(KDp[j], Ub, Db, co, co, M);
        combine(Ub, Db, Ub, co, co, M / 3);
        cevn(Ub, co, nc * KNN_);
        attention_kernel<<<dim3((unsigned)nc, BSZ), 256, 0, stream>>>(
            Ub, QF, Tb, outF, co, nc, n0, ndst);
      }
    }
    if (i >= 2) {
      const int j = i - 2;
      mean_n_kernel<<<grid1(BSZ * co * 3), 256, 0, stream>>>(outF, GB, co,
                                                             ndst);
      concat_kernel<<<grid1(BSZ * 2 * co * 3 * ndst), 256, 0, stream>>>(
          outF, GB, FC, co, ndst);
      vec_lna_dense(GLp[j], GDp[j], FC, outF, co, co, 2 * co, 3 * ndst);
    }
    curF = outF;
    curX = dX;
    f_i ^= 1;
  }

  // ---- head ----
  vec_lna_dense(CW, CD, curF, FF, 256, 1, 512, 96);
  mean_n_kernel<<<grid1(BSZ * 256 * 3), 256, 0, stream>>>(FF, FM, 256, 32);
  copy_kernel<<<grid1(BSZ * 256 * 3), 256, 0, stream>>>(FM, ZS, BSZ * 256 * 3);
  cevn(ZS, 256, 1);
  gemm(FCW, FM, ZI, 256, 256, 3);
  cevn(ZI, 256, 1);
  float* out = (float*)d_out;
  scale_out_kernel<<<1, 32, 0, stream>>>(FM, out);
  write_final_kernel<<<grid1(BSZ * 256), 256, 0, stream>>>(ZS, ZI, out);
}